// GraphDegreeConv_56710748176714
// MI455X (gfx1250) — compile-verified
//
#include <hip/hip_runtime.h>
#include <hip/hip_bf16.h>

typedef float v2f __attribute__((ext_vector_type(2)));
typedef float v8f __attribute__((ext_vector_type(8)));

#define NODE   128
#define EDGE   64
#define OUTC   128
#define KCAT   320            // 128 (atom sum) + 64 (bond sum) + 128 (self)
#define LDW    324            // padded row stride (324 mod 64 banks -> conflict-free)
#define NPD    125000
#define NATOMS 500000
#define NTILES ((NPD + 15) / 16)   // 7813 (last tile has 8 valid rows)
#define BN_EPS 1e-5f

// ---------------------------------------------------------------------------
// Phase 0: zero BN accumulators in workspace
// ws layout (floats): [0,128)=sum  [128,256)=sumsq  [256,384)=mean  [384,512)=scale
// ---------------------------------------------------------------------------
__global__ void gdc_zero_stats(float* stats) {
    if (threadIdx.x < 512) stats[threadIdx.x] = 0.0f;
}

// ---------------------------------------------------------------------------
// Phase 1: fused gather-sum + (neighbor|self) GEMM via v_wmma_f32_16x16x4_f32
// grid = (BLOCKS_X, 4), block = 256 (8 waves of 32); blockIdx.y = degree-1
// ---------------------------------------------------------------------------
__global__ __launch_bounds__(256) void gdc_fused_kernel(
    const float* __restrict__ atom_repr, const float* __restrict__ bond_repr,
    const int* __restrict__ ai1, const int* __restrict__ bi1, const float* __restrict__ W1,
    const int* __restrict__ ai2, const int* __restrict__ bi2, const float* __restrict__ W2,
    const int* __restrict__ ai3, const int* __restrict__ bi3, const float* __restrict__ W3,
    const int* __restrict__ ai4, const int* __restrict__ bi4, const float* __restrict__ W4,
    const float* __restrict__ W_self, const float* __restrict__ bias,
    float* __restrict__ xout, float* __restrict__ stats)
{
    extern __shared__ float lds[];
    float* Wl   = lds;                       // 128 * LDW floats  (B = [W_d | W_self]^T view)
    float* Al   = lds + 128 * LDW;           // 16  * LDW floats  (A tile)
    float* sSum = Al  + 16  * LDW;           // 128
    float* sSq  = sSum + 128;                // 128

    const int tid = threadIdx.x;
    const int deg = blockIdx.y;              // 0..3
    const int d   = deg + 1;
    const int rowOffset = deg * NPD;

    const int* aidx; const int* bidx; const float* Wd;
    if      (deg == 0) { aidx = ai1; bidx = bi1; Wd = W1; }
    else if (deg == 1) { aidx = ai2; bidx = bi2; Wd = W2; }
    else if (deg == 2) { aidx = ai3; bidx = bi3; Wd = W3; }
    else               { aidx = ai4; bidx = bi4; Wd = W4; }

    if (tid < 128) { sSum[tid] = 0.0f; sSq[tid] = 0.0f; }

    // Stage concatenated weight matrix: Wl[n*LDW + k] = k<192 ? Wd[n][k] : W_self[n][k-192]
    for (int i = tid; i < (128 * 192) / 4; i += 256) {
        const int n = (i * 4) / 192, k = (i * 4) % 192;
        float4 v = ((const float4*)Wd)[i];
        *(float4*)&Wl[n * LDW + k] = v;
    }
    for (int i = tid; i < (128 * 128) / 4; i += 256) {
        const int n = (i * 4) / 128, k = (i * 4) % 128;
        float4 v = ((const float4*)W_self)[i];
        *(float4*)&Wl[n * LDW + 192 + k] = v;
    }
    __syncthreads();

    const int lane  = tid & 31;
    const int wave  = tid >> 5;
    const int n0    = wave * 16;             // this wave's N-tile (8 waves cover 128 cols)
    const int m16   = lane & 15;
    const int halfK = (lane < 16) ? 0 : 2;   // ISA 16x4 A / 4x16 B lane split
    const int mBase = (lane < 16) ? 0 : 8;   // ISA 16x16 C/D lane split
    const float bval = bias[n0 + m16];

    for (int tile = blockIdx.x; tile < NTILES; tile += gridDim.x) {
        const int jbase  = tile * 16;
        const int nvalid = min(16, NPD - jbase);

        // ---- stage A tile: 16 rows x [a_sum(128) | b_sum(64) | self(128)] ----
        {
            const int r   = tid >> 4;        // 0..15 (row in tile)
            const int c16 = tid & 15;        // 0..15 (column sub-slice)
            const int j   = jbase + r;
            const bool ok = (r < nvalid);
            float asum[8] = {0,0,0,0,0,0,0,0};
            float bsum[4] = {0,0,0,0};
            float selfv[8] = {0,0,0,0,0,0,0,0};
            if (ok) {
                for (int k = 0; k < d; k++) {
                    const float* ap = atom_repr + (size_t)aidx[j * d + k] * NODE + c16 * 8;
                    #pragma unroll
                    for (int t = 0; t < 8; t++) asum[t] += ap[t];
                    const float* bp = bond_repr + (size_t)bidx[j * d + k] * EDGE + c16 * 4;
                    #pragma unroll
                    for (int t = 0; t < 4; t++) bsum[t] += bp[t];
                }
                const float* sp = atom_repr + (size_t)(rowOffset + j) * NODE + c16 * 8;
                #pragma unroll
                for (int t = 0; t < 8; t++) selfv[t] = sp[t];
            }
            float* arow = &Al[r * LDW];
            #pragma unroll
            for (int t = 0; t < 8; t++) arow[c16 * 8 + t] = asum[t];
            #pragma unroll
            for (int t = 0; t < 4; t++) arow[128 + c16 * 4 + t] = bsum[t];
            #pragma unroll
            for (int t = 0; t < 8; t++) arow[192 + c16 * 8 + t] = selfv[t];
        }
        __syncthreads();

        // ---- K = 320 accumulation: 80 x v_wmma_f32_16x16x4_f32 ----
        v8f acc = {0.f,0.f,0.f,0.f,0.f,0.f,0.f,0.f};
        const float* ap = &Al[m16 * LDW + halfK];               // A[m16][kb + halfK..+1]
        const float* wp = &Wl[(n0 + m16) * LDW + halfK];        // B[kb+halfK..+1][n0+m16] = Wcat[n][k]
        #pragma unroll 8
        for (int kb = 0; kb < KCAT; kb += 4) {
            v2f a = *(const v2f*)(ap + kb);
            v2f b = *(const v2f*)(wp + kb);
            acc = __builtin_amdgcn_wmma_f32_16x16x4_f32(
                false, a, false, b, (short)0, acc, false, false);
        }

        // ---- bias, store x, BN partial stats ----
        float s = 0.0f, sq = 0.0f;
        float* gp = xout + ((size_t)(rowOffset + jbase + mBase)) * OUTC + n0 + m16;
        if (nvalid == 16) {
            // Fast path (all tiles but the last): unconditional stores, base + imm offsets
            #pragma unroll
            for (int v = 0; v < 8; v++) {
                const float val = acc[v] + bval;
                gp[(size_t)v * OUTC] = val;
                s += val; sq += val * val;
            }
        } else {
            // Ragged last tile (8 valid rows)
            #pragma unroll
            for (int v = 0; v < 8; v++) {
                const int M = mBase + v;                        // D layout: VGPR v -> row v / v+8
                const float val = acc[v] + bval;
                if (M < nvalid) {
                    gp[(size_t)v * OUTC] = val;
                    s += val; sq += val * val;
                }
            }
        }
        atomicAdd(&sSum[n0 + m16], s);                          // ds_add_f32
        atomicAdd(&sSq [n0 + m16], sq);
        __syncthreads();
    }

    __syncthreads();
    if (tid < 128) {
        atomicAdd(&stats[tid], sSum[tid]);                      // global_atomic_add_f32
        atomicAdd(&stats[128 + tid], sSq[tid]);
    }
}

// ---------------------------------------------------------------------------
// Phase 2: finalize BN statistics (biased batch stats, training mode)
// ---------------------------------------------------------------------------
__global__ void gdc_bn_finalize(float* stats) {
    const int c = threadIdx.x;
    if (c < 128) {
        const float inv_n = 1.0f / (float)NATOMS;
        const float mean  = stats[c] * inv_n;
        const float var   = stats[128 + c] * inv_n - mean * mean;
        stats[256 + c] = mean;
        stats[384 + c] = rsqrtf(var + BN_EPS);
    }
}

// ---------------------------------------------------------------------------
// Phase 3: in-place normalize + ReLU (float4 vectorized, 16M vec4 elements)
// ---------------------------------------------------------------------------
__global__ __launch_bounds__(256) void gdc_bn_relu(float* __restrict__ x,
                                                   const float* __restrict__ stats) {
    const size_t i = (size_t)blockIdx.x * 256 + threadIdx.x;    // vec4 index
    const int c = (int)((i * 4) & (OUTC - 1));                  // channel of first elem
    float4 v = ((float4*)x)[i];
    v.x = fmaxf((v.x - stats[256 + c + 0]) * stats[384 + c + 0], 0.0f);
    v.y = fmaxf((v.y - stats[256 + c + 1]) * stats[384 + c + 1], 0.0f);
    v.z = fmaxf((v.z - stats[256 + c + 2]) * stats[384 + c + 2], 0.0f);
    v.w = fmaxf((v.w - stats[256 + c + 3]) * stats[384 + c + 3], 0.0f);
    ((float4*)x)[i] = v;
}

// ---------------------------------------------------------------------------
extern "C" void kernel_launch(void* const* d_in, const int* in_sizes, int n_in,
                              void* d_out, int out_size, void* d_ws, size_t ws_size,
                              hipStream_t stream) {
    const float* atom_repr = (const float*)d_in[0];
    const float* bond_repr = (const float*)d_in[1];
    const int*   ai1 = (const int*)d_in[2];
    const int*   bi1 = (const int*)d_in[3];
    const float* W1  = (const float*)d_in[4];
    const int*   ai2 = (const int*)d_in[5];
    const int*   bi2 = (const int*)d_in[6];
    const float* W2  = (const float*)d_in[7];
    const int*   ai3 = (const int*)d_in[8];
    const int*   bi3 = (const int*)d_in[9];
    const float* W3  = (const float*)d_in[10];
    const int*   ai4 = (const int*)d_in[11];
    const int*   bi4 = (const int*)d_in[12];
    const float* W4  = (const float*)d_in[13];
    const float* W_self = (const float*)d_in[14];
    const float* bias   = (const float*)d_in[15];
    float* out   = (float*)d_out;
    float* stats = (float*)d_ws;

    gdc_zero_stats<<<1, 512, 0, stream>>>(stats);

    const size_t ldsBytes = (size_t)(128 * LDW + 16 * LDW + 256) * sizeof(float); // 187,648 B
    dim3 grid(256, 4);
    gdc_fused_kernel<<<grid, 256, ldsBytes, stream>>>(
        atom_repr, bond_repr,
        ai1, bi1, W1, ai2, bi2, W2, ai3, bi3, W3, ai4, bi4, W4,
        W_self, bias, out, stats);

    gdc_bn_finalize<<<1, 128, 0, stream>>>(stats);

    const int nvec4 = (NATOMS * OUTC) / 4;          // 16,000,000
    gdc_bn_relu<<<nvec4 / 256, 256, 0, stream>>>(out, stats);
}